// ChamferLoss_6863357739534
// MI455X (gfx1250) — compile-verified
//
#include <hip/hip_runtime.h>
#include <math.h>

typedef __attribute__((ext_vector_type(2))) float v2f;
typedef __attribute__((ext_vector_type(8))) float v8f;

#define NPTS 8192
#define NBATCH 4
#define TILE 16
#define NTILES (NPTS / TILE)            // 512 column tiles
#define WAVES 8
#define ROWS_PER_BLOCK (WAVES * TILE)   // 128 rows per 256-thread block

// Region layout in the prep buffer: 8 regions indexed r = d*NBATCH + b
// (d=0: columns come from label, d=1: columns come from pred).
// Each region: NTILES tiles * 32 lanes * v2f. Lane l of tile t holds the
// WMMA B-operand register pair for column j = t*16 + (l&15):
//   lanes 0-15  (K=0,1): { -2*qx, -2*qy }
//   lanes 16-31 (K=2,3): { -2*qz,  y2  }   (y2 = qx^2+qy^2+qz^2)
// With A's K3 slot set to 1.0, the WMMA produces  c = y2 - 2*x.y  directly.
__global__ __launch_bounds__(256)
void chamfer_prep(const float* __restrict__ pred,
                  const float* __restrict__ label,
                  v2f* __restrict__ prep)
{
    const unsigned u = blockIdx.x * 256u + threadIdx.x;  // 0 .. 2*4*512*32-1
    const int lane = u & 31;
    const int t    = (u >> 5) & (NTILES - 1);
    const int b    = (u >> 14) & (NBATCH - 1);
    const int d    = (u >> 16) & 1;

    const int j = t * TILE + (lane & 15);
    const float* src = (d ? pred : label) + ((size_t)b * NPTS + j) * 3;
    const float qx = src[0], qy = src[1], qz = src[2];
    const float y2 = qx * qx + qy * qy + qz * qz;

    v2f o;
    if (lane < 16) { o.x = -2.0f * qx; o.y = -2.0f * qy; }
    else           { o.x = -2.0f * qz; o.y = y2; }
    prep[u] = o;   // coalesced b64 store
}

// One-directional chamfer row-min. A tile = 16 row points (K=3 padded with
// 1.0), B operands pre-baked by chamfer_prep. Inner loop per 16x16 tile:
// one global_load_b64 + one v_wmma_f32_16x16x4_f32 + mins.
__global__ __launch_bounds__(256)
void chamfer_rowmin_wmma(const float* __restrict__ P,
                         const v2f* __restrict__ prepDir,  // direction base
                         float* __restrict__ rowdist)
{
    const int b = blockIdx.y;
    P += (size_t)b * NPTS * 3;
    rowdist += (size_t)b * NPTS;

    const int tid  = threadIdx.x;
    const int lane = tid & 31;
    const int wave = tid >> 5;
    const int h    = lane >> 4;   // 16-lane half
    const int n    = lane & 15;

    const v2f* qp = prepDir + (size_t)b * (NTILES * 32) + lane;

    const int rowBase = blockIdx.x * ROWS_PER_BLOCK + wave * TILE;
    const int row = rowBase + n;

    // A tile 16x4 f32: lanes 0-15 -> {x,y} (K0,K1); lanes 16-31 -> {z,1} (K2,K3).
    const float px = P[row * 3 + 0];
    const float py = P[row * 3 + 1];
    const float pz = P[row * 3 + 2];
    const float x2 = px * px + py * py + pz * pz;

    v2f a;
    a.x = h ? pz : px;
    a.y = h ? 1.0f : py;

    // Loop-invariant broadcast: x2 of output-row (v + 8*h) for each C VGPR v.
    float rowx2[8];
#pragma unroll
    for (int v = 0; v < 8; ++v)
        rowx2[v] = __shfl(x2, v + 8 * h, 32);

    v8f minrow;
#pragma unroll
    for (int v = 0; v < 8; ++v)
        minrow[v] = __builtin_inff();

    for (int t = 0; t < NTILES; ++t) {
        const v2f bb = qp[(size_t)t * 32];   // coalesced 256B/wave
        v8f c = {};
        c = __builtin_amdgcn_wmma_f32_16x16x4_f32(
                /*neg_a=*/false, a, /*neg_b=*/false, bb,
                /*c_mod=*/(short)0, c, /*reuse_a=*/false, /*reuse_b=*/false);
        // c[v] = y2 - 2*x.y for (row v+8h, col n); just min it.
#pragma unroll
        for (int v = 0; v < 8; ++v)
            minrow[v] = fminf(minrow[v], c[v]);
    }

    // Min across the 16 columns (lanes of each half), then finish rows.
#pragma unroll
    for (int v = 0; v < 8; ++v) {
        float m = minrow[v];
        m = fminf(m, __shfl_xor(m, 1, 32));
        m = fminf(m, __shfl_xor(m, 2, 32));
        m = fminf(m, __shfl_xor(m, 4, 32));
        m = fminf(m, __shfl_xor(m, 8, 32));
        m += rowx2[v];               // restore hoisted ||x||^2 (exact)
        m = fmaxf(m, 0.0f);          // guard rounding-induced negatives
        const float dgt = sqrtf(m);
        if (n == 0)                  // lane 0 / lane 16 write their 8 rows
            rowdist[rowBase + v + 8 * h] = dgt;
    }
}

// Deterministic final reduction: fixed strided per-thread order + LDS tree.
__global__ __launch_bounds__(256)
void chamfer_final_reduce(const float* __restrict__ rowdist,
                          float* __restrict__ out)
{
    __shared__ float smem[256];
    float s = 0.0f;
    for (int i = threadIdx.x; i < 2 * NBATCH * NPTS; i += 256)
        s += rowdist[i];
    smem[threadIdx.x] = s;
    __syncthreads();
    for (int off = 128; off > 0; off >>= 1) {
        if (threadIdx.x < off)
            smem[threadIdx.x] += smem[threadIdx.x + off];
        __syncthreads();
    }
    if (threadIdx.x == 0)
        out[0] = smem[0] / (float)(NBATCH * NPTS);  // (S_xy + S_yx) / 32768
}

extern "C" void kernel_launch(void* const* d_in, const int* in_sizes, int n_in,
                              void* d_out, int out_size, void* d_ws, size_t ws_size,
                              hipStream_t stream) {
    const float* pred  = (const float*)d_in[0];
    const float* label = (const float*)d_in[1];

    float* rowdist = (float*)d_ws;                       // 65536 floats (256 KB)
    v2f*   prep    = (v2f*)(rowdist + 2 * NBATCH * NPTS); // 131072 v2f (1 MB)

    // Bake B operands for both directions (d=0: label cols, d=1: pred cols).
    chamfer_prep<<<(2 * NBATCH * NTILES * 32) / 256, 256, 0, stream>>>(
        pred, label, prep);

    dim3 grid(NPTS / ROWS_PER_BLOCK, NBATCH);
    const size_t regionStride = (size_t)NBATCH * NTILES * 32;  // v2f per direction
    // pred rows vs label cols
    chamfer_rowmin_wmma<<<grid, 256, 0, stream>>>(pred,  prep,                rowdist);
    // label rows vs pred cols
    chamfer_rowmin_wmma<<<grid, 256, 0, stream>>>(label, prep + regionStride, rowdist + NBATCH * NPTS);

    chamfer_final_reduce<<<1, 256, 0, stream>>>(rowdist, (float*)d_out);
}